// DiffRenderer_64355789964092
// MI455X (gfx1250) — compile-verified
//
#include <hip/hip_runtime.h>

#define GD 96
#define GD3 (GD*GD*GD)
#define W0 640
#define H0 480
#define HW (W0*H0)
#define NV 9
#define VOXEL 0.04f
#define DEPTH_MIN_C 0.025f   // 0.001/0.04
#define DEPTH_MAX_C 100.0f   // 4.0/0.04
#define BIGF 1e30f

typedef float v2f __attribute__((ext_vector_type(2)));
typedef float v8f __attribute__((ext_vector_type(8)));

// ---------------- init ----------------
__global__ void init_k(unsigned* __restrict__ dflatU,
                       unsigned* __restrict__ dminb, unsigned* __restrict__ dmaxb,
                       unsigned* __restrict__ tminb, unsigned* __restrict__ tmaxb) {
  int idx = blockIdx.x * blockDim.x + threadIdx.x;
  if (idx < NV * HW) dflatU[idx] = __float_as_uint(BIGF);
  if (idx < NV) {
    dminb[idx] = __float_as_uint(BIGF);
    dmaxb[idx] = 0u;                       // bits(0.0f); all candidates >= 0
    tminb[idx] = __float_as_uint(BIGF);
    tmaxb[idx] = 0u;
  }
}

// ---------------- deterministic sparse scatter (highest index wins) ----------------
__global__ void scatter_winner_k(const int* __restrict__ coords, int* __restrict__ winner, int N) {
  int i = blockIdx.x * blockDim.x + threadIdx.x;
  if (i >= N) return;
  int l0 = min(max(coords[4 * i + 3], 0), GD - 1);
  int l1 = min(max(coords[4 * i + 2], 0), GD - 1);
  int l2 = min(max(coords[4 * i + 1], 0), GD - 1);
  atomicMax(&winner[(l0 * GD + l1) * GD + l2], i);
}

__global__ void scatter_val_k(const int* __restrict__ coords, const float* __restrict__ sdf,
                              const int* __restrict__ winner, float* __restrict__ grid, int N) {
  int i = blockIdx.x * blockDim.x + threadIdx.x;
  if (i >= N) return;
  int l0 = min(max(coords[4 * i + 3], 0), GD - 1);
  int l1 = min(max(coords[4 * i + 2], 0), GD - 1);
  int l2 = min(max(coords[4 * i + 1], 0), GD - 1);
  int cell = (l0 * GD + l1) * GD + l2;
  if (winner[cell] == i) grid[cell] = sdf[i];
}

// ---------------- sparse central-difference normals ----------------
__global__ void normals_k(const int* __restrict__ coords, const float* __restrict__ grid,
                          float* __restrict__ nraw, int N) {
  int i = blockIdx.x * blockDim.x + threadIdx.x;
  if (i >= N) return;
  int l0 = min(max(coords[4 * i + 3], 0), GD - 1);
  int l1 = min(max(coords[4 * i + 2], 0), GD - 1);
  int l2 = min(max(coords[4 * i + 1], 0), GD - 1);
  float nx = 0.f, ny = 0.f, nz = 0.f;
  if (l0 >= 1 && l0 < GD - 1 && l1 >= 1 && l1 < GD - 1 && l2 >= 1 && l2 < GD - 1) {
    int b = (l0 * GD + l1) * GD + l2;
    nx = grid[b + 1] - grid[b - 1];
    ny = grid[b + GD] - grid[b - GD];
    nz = grid[b + GD * GD] - grid[b - GD * GD];
  }
  nraw[3 * i + 0] = nx;
  nraw[3 * i + 1] = ny;
  nraw[3 * i + 2] = nz;
}

// ---------------- WMMA 16x16x4 f32 transform: pc = [R|t]*(p,1), Rn = R*n ----------------
// A (16x4): rows 0..2 = [R | t], rows 3..5 = [R | 0], rest 0.
// A layout: lane m (<16) holds (A[m][0], A[m][1]); lane m+16 holds (A[m][2], A[m][3]).
// B (4x16): column n = (x,y,z,1) of point n (or (nx,ny,nz,0)).
// B layout: lane n (<16) = (B[0][n], B[1][n]); lane n+16 = (B[2][n], B[3][n]).
// D: element j at lane n(<16) = D[j][n]  ->  j=0..2 are the transformed components.
__device__ __forceinline__ void transform32(
    int lane, float px, float py, float pz, float nx, float ny, float nz,
    const float* __restrict__ vmv,
    float& pcx, float& pcy, float& pcz, float& rnx, float& rny, float& rnz) {
  int m = lane & 15;
  bool hi = lane >= 16;

  float a0, a1;
  if (m < 3) {
    a0 = vmv[m * 4 + (hi ? 2 : 0)];
    a1 = vmv[m * 4 + (hi ? 3 : 1)];
  } else if (m < 6) {
    int r = m - 3;
    a0 = vmv[r * 4 + (hi ? 2 : 0)];
    a1 = hi ? 0.0f : vmv[r * 4 + 1];
  } else {
    a0 = 0.0f;
    a1 = 0.0f;
  }
  v2f A; A[0] = a0; A[1] = a1;

  // gather operands for the split-K B layout (convergent shuffles)
  float zl  = __shfl(pz, m);
  float xh  = __shfl(px, m + 16);
  float yh  = __shfl(py, m + 16);
  float nzl = __shfl(nz, m);
  float nxh = __shfl(nx, m + 16);
  float nyh = __shfl(ny, m + 16);

  v2f B1p; B1p[0] = hi ? zl  : px;  B1p[1] = hi ? 1.0f : py;   // points 0..15
  v2f B2p; B2p[0] = hi ? pz  : xh;  B2p[1] = hi ? 1.0f : yh;   // points 16..31
  v2f B1n; B1n[0] = hi ? nzl : nx;  B1n[1] = hi ? 0.0f : ny;   // normals 0..15
  v2f B2n; B2n[0] = hi ? nz  : nxh; B2n[1] = hi ? 0.0f : nyh;  // normals 16..31

  v8f c0 = {0.f, 0.f, 0.f, 0.f, 0.f, 0.f, 0.f, 0.f};
  v8f d1 = __builtin_amdgcn_wmma_f32_16x16x4_f32(false, A, false, B1p, (short)0, c0, false, false);
  v8f d2 = __builtin_amdgcn_wmma_f32_16x16x4_f32(false, A, false, B2p, (short)0, c0, false, false);
  v8f e1 = __builtin_amdgcn_wmma_f32_16x16x4_f32(false, A, false, B1n, (short)0, c0, false, false);
  v8f e2 = __builtin_amdgcn_wmma_f32_16x16x4_f32(false, A, false, B2n, (short)0, c0, false, false);

  // redistribute so every lane holds its own point's results
  float sx = __shfl(d2[0], m), sy = __shfl(d2[1], m), sz = __shfl(d2[2], m);
  float tx = __shfl(e2[0], m), ty = __shfl(e2[1], m), tz = __shfl(e2[2], m);
  pcx = hi ? sx : d1[0];
  pcy = hi ? sy : d1[1];
  pcz = hi ? sz : d1[2];
  rnx = hi ? tx : e1[0];
  rny = hi ? ty : e1[1];
  rnz = hi ? tz : e1[2];
}

__device__ __forceinline__ void project(float pcx, float pcy, float pcz,
                                        const float* __restrict__ intrv,
                                        float& z, int& pix, bool& valid) {
  z = pcz;
  float zs = (fabsf(z) < 1e-6f) ? 1e-6f : z;
  float rz = 1.0f / zs;                               // single divide, shared by u and v
  float uf = rintf(intrv[0] * pcx * rz + intrv[2]);
  float vf = rintf(intrv[5] * pcy * rz + intrv[6]);
  int ui = (int)uf, vi = (int)vf;
  valid = (z > DEPTH_MIN_C) && (z < DEPTH_MAX_C) &&
          (ui >= 0) && (ui < W0) && (vi >= 0) && (vi < H0);
  int uc = min(max(ui, 0), W0 - 1);
  int vc = min(max(vi, 0), H0 - 1);
  pix = vc * W0 + uc;
}

__device__ __forceinline__ void load_point(const int* __restrict__ coords,
                                           const float* __restrict__ origin,
                                           const float* __restrict__ nraw,
                                           int ic, float& px, float& py, float& pz,
                                           float& nx, float& ny, float& nz) {
  px = (float)coords[4 * ic + 3] + origin[0];
  py = (float)coords[4 * ic + 2] + origin[1];
  pz = (float)coords[4 * ic + 1] + origin[2];
  nx = nraw[3 * ic + 0];
  ny = nraw[3 * ic + 1];
  nz = nraw[3 * ic + 2];
}

// ---------------- pass 1: z-buffer atomic min ----------------
__global__ void proj_min_k(const int* __restrict__ coords, const float* __restrict__ origin,
                           const float* __restrict__ nraw, const float* __restrict__ vm,
                           const float* __restrict__ intr, unsigned* __restrict__ dflatU, int N) {
  int i = blockIdx.x * blockDim.x + threadIdx.x;
  int v = blockIdx.y;
  int ic = min(i, N - 1);
  bool active = (i < N);
  float px, py, pz, nx, ny, nz;
  load_point(coords, origin, nraw, ic, px, py, pz, nx, ny, nz);
  float pcx, pcy, pcz, rnx, rny, rnz;
  transform32(threadIdx.x & 31, px, py, pz, nx, ny, nz, vm + v * 16,
              pcx, pcy, pcz, rnx, rny, rnz);
  float z; int pix; bool valid;
  project(pcx, pcy, pcz, intr + v * 16, z, pix, valid);
  if (active && valid)
    atomicMin(&dflatU[v * HW + pix], __float_as_uint(z));  // z > 0: bit order == float order
}

// ---------------- pass 2: winner normal accumulation ----------------
__global__ void proj_acc_k(const int* __restrict__ coords, const float* __restrict__ origin,
                           const float* __restrict__ nraw, const float* __restrict__ vm,
                           const float* __restrict__ intr, const unsigned* __restrict__ dflatU,
                           float* __restrict__ nsum, float* __restrict__ ncnt, int N) {
  int i = blockIdx.x * blockDim.x + threadIdx.x;
  int v = blockIdx.y;
  int ic = min(i, N - 1);
  bool active = (i < N);
  float px, py, pz, nx, ny, nz;
  load_point(coords, origin, nraw, ic, px, py, pz, nx, ny, nz);
  float pcx, pcy, pcz, rnx, rny, rnz;
  transform32(threadIdx.x & 31, px, py, pz, nx, ny, nz, vm + v * 16,
              pcx, pcy, pcz, rnx, rny, rnz);
  float z; int pix; bool valid;
  project(pcx, pcy, pcz, intr + v * 16, z, pix, valid);
  if (active && valid) {
    float df = __uint_as_float(dflatU[v * HW + pix]);
    if (z <= df + 1e-6f) {
      float nlen = sqrtf(rnx * rnx + rny * rny + rnz * rnz);
      float s = -1.0f / fmaxf(nlen, 1e-5f);
      long b = ((long)v * HW + pix) * 3;
      atomicAdd(&nsum[b + 0], rnx * s);
      atomicAdd(&nsum[b + 1], rny * s);
      atomicAdd(&nsum[b + 2], rnz * s);
      atomicAdd(&ncnt[v * HW + pix], 1.0f);
    }
  }
}

// ---------------- wave32 butterfly reductions + small cross-wave combine ----------------
__device__ __forceinline__ float wred_sum(float x) {
  for (int o = 16; o > 0; o >>= 1) x += __shfl_xor(x, o);
  return x;
}
__device__ __forceinline__ float wred_min(float x) {
  for (int o = 16; o > 0; o >>= 1) x = fminf(x, __shfl_xor(x, o));
  return x;
}
__device__ __forceinline__ float wred_max(float x) {
  for (int o = 16; o > 0; o >>= 1) x = fmaxf(x, __shfl_xor(x, o));
  return x;
}
// blockDim.x == 256 -> 8 waves
__device__ __forceinline__ float bred_sum(float x, float* sm) {
  x = wred_sum(x);
  __syncthreads();
  if ((threadIdx.x & 31) == 0) sm[threadIdx.x >> 5] = x;
  __syncthreads();
  float r = sm[0];
  #pragma unroll
  for (int k = 1; k < 8; ++k) r += sm[k];
  return r;
}
__device__ __forceinline__ float bred_min(float x, float* sm) {
  x = wred_min(x);
  __syncthreads();
  if ((threadIdx.x & 31) == 0) sm[threadIdx.x >> 5] = x;
  __syncthreads();
  float r = sm[0];
  #pragma unroll
  for (int k = 1; k < 8; ++k) r = fminf(r, sm[k]);
  return r;
}
__device__ __forceinline__ float bred_max(float x, float* sm) {
  x = wred_max(x);
  __syncthreads();
  if ((threadIdx.x & 31) == 0) sm[threadIdx.x >> 5] = x;
  __syncthreads();
  float r = sm[0];
  #pragma unroll
  for (int k = 1; k < 8; ++k) r = fmaxf(r, sm[k]);
  return r;
}

// ---------------- finalize depth/normal images + min/max/count reductions ----------------
__global__ void finalize_k(unsigned* __restrict__ dflatU, const float* __restrict__ ncnt,
                           float* __restrict__ nsum, const float* __restrict__ dt,
                           unsigned* __restrict__ dminb, unsigned* __restrict__ dmaxb,
                           unsigned* __restrict__ tminb, unsigned* __restrict__ tmaxb,
                           float* __restrict__ cnt) {
  __shared__ float sm[8];
  int p = blockIdx.x * blockDim.x + threadIdx.x;
  int v = blockIdx.y;
  long idx = (long)v * HW + p;
  float df = __uint_as_float(dflatU[idx]);
  bool hit = df < 0.5f * BIGF;
  float dimg = hit ? df * VOXEL : 0.0f;
  ((float*)dflatU)[idx] = dimg;                       // overwrite z-bits with depth image
  float inv = hit ? 1.0f / fmaxf(ncnt[idx], 1.0f) : 0.0f;
  nsum[idx * 3 + 0] *= inv;
  nsum[idx * 3 + 1] *= inv;
  nsum[idx * 3 + 2] *= inv;
  float tv = dt[idx];
  float bmin = bred_min(dimg, sm);
  float bmax = bred_max(dimg, sm);
  float tmin = bred_min(tv, sm);
  float tmax = bred_max(tv, sm);
  float bc   = bred_sum((tv != 0.0f) ? 1.0f : 0.0f, sm);
  if (threadIdx.x == 0) {
    atomicMin(&dminb[v], __float_as_uint(bmin));       // all values >= 0
    atomicMax(&dmaxb[v], __float_as_uint(bmax));
    atomicMin(&tminb[v], __float_as_uint(tmin));
    atomicMax(&tmaxb[v], __float_as_uint(tmax));
    atomicAdd(&cnt[v], bc);
  }
}

// ---------------- normalize01 + L1 partials ----------------
__global__ void norm_loss_k(float* __restrict__ depths, const float* __restrict__ dt,
                            float* __restrict__ dtgt,
                            const unsigned* __restrict__ dminb, const unsigned* __restrict__ dmaxb,
                            const unsigned* __restrict__ tminb, const unsigned* __restrict__ tmaxb,
                            float* __restrict__ lsum) {
  __shared__ float sm[8];
  int p = blockIdx.x * blockDim.x + threadIdx.x;
  int v = blockIdx.y;
  long idx = (long)v * HW + p;
  float dmin = __uint_as_float(dminb[v]), dmax = __uint_as_float(dmaxb[v]);
  float tmin = __uint_as_float(tminb[v]), tmax = __uint_as_float(tmaxb[v]);
  float dd = dmax - dmin, td = tmax - tmin;
  float dn = (dd == 0.0f) ? 0.0f : (depths[idx] - dmin) / dd;
  depths[idx] = dn;
  float tv = dt[idx];
  float tn = (td == 0.0f) ? 0.0f : (tv - tmin) / td;
  dtgt[idx] = tn;
  float c = (tv != 0.0f) ? fabsf(dn - tn) : 0.0f;
  float bc = bred_sum(c, sm);
  if (threadIdx.x == 0) atomicAdd(&lsum[v], bc);
}

__global__ void loss_k(const float* __restrict__ lsum, const float* __restrict__ cnt,
                       float* __restrict__ loss) {
  if (blockIdx.x == 0 && threadIdx.x == 0) {
    float acc = 0.0f;
    for (int v = 0; v < NV; ++v) {
      float c = cnt[v];
      float l = (c > 0.0f) ? lsum[v] / fmaxf(c, 1.0f) : 0.0f;
      acc += l * (1.0f / (float)NV);
    }
    loss[0] = acc;
  }
}

extern "C" void kernel_launch(void* const* d_in, const int* in_sizes, int n_in,
                              void* d_out, int out_size, void* d_ws, size_t ws_size,
                              hipStream_t stream) {
  const int*   coords  = (const int*)d_in[0];
  const float* origin  = (const float*)d_in[1];
  const float* sdf     = (const float*)d_in[2];
  const float* dt      = (const float*)d_in[4];
  const float* intr    = (const float*)d_in[5];
  const float* vm      = (const float*)d_in[6];
  int N = in_sizes[0] / 4;

  // workspace layout
  char* ws = (char*)d_ws;
  size_t o = 0;
  int*      winner = (int*)(ws + o);      o += (size_t)GD3 * 4;
  float*    grid   = (float*)(ws + o);    o += (size_t)GD3 * 4;
  float*    nraw   = (float*)(ws + o);    o += (size_t)N * 3 * 4;
  float*    ncnt   = (float*)(ws + o);    o += (size_t)NV * HW * 4;
  unsigned* dminb  = (unsigned*)(ws + o); o += NV * 4;
  unsigned* dmaxb  = (unsigned*)(ws + o); o += NV * 4;
  unsigned* tminb  = (unsigned*)(ws + o); o += NV * 4;
  unsigned* tmaxb  = (unsigned*)(ws + o); o += NV * 4;
  float*    cnt    = (float*)(ws + o);    o += NV * 4;
  float*    lsum   = (float*)(ws + o);    o += NV * 4;

  // output layout: [loss | depths(9*HW) | dtgt(9*HW) | normals(9*HW*3)]
  float* out    = (float*)d_out;
  float* loss   = out;
  float* depths = out + 1;
  float* dtgt   = out + 1 + (size_t)NV * HW;
  float* norms  = out + 1 + (size_t)2 * NV * HW;
  unsigned* dflatU = (unsigned*)depths;   // z-buffer bits live in the depths region

  hipMemsetAsync(winner, 0xFF, (size_t)GD3 * 4, stream);        // winner = -1
  hipMemsetAsync(grid, 0, (size_t)GD3 * 4, stream);
  hipMemsetAsync(ncnt, 0, (size_t)NV * HW * 4, stream);
  hipMemsetAsync(norms, 0, (size_t)NV * HW * 3 * 4, stream);
  hipMemsetAsync(cnt, 0, (size_t)NV * 2 * 4, stream);           // cnt + lsum contiguous

  int nb = (N + 255) / 256;
  init_k<<<(NV * HW + 255) / 256, 256, 0, stream>>>(dflatU, dminb, dmaxb, tminb, tmaxb);
  scatter_winner_k<<<nb, 256, 0, stream>>>(coords, winner, N);
  scatter_val_k<<<nb, 256, 0, stream>>>(coords, sdf, winner, grid, N);
  normals_k<<<nb, 256, 0, stream>>>(coords, grid, nraw, N);

  dim3 gp(nb, NV);
  proj_min_k<<<gp, 256, 0, stream>>>(coords, origin, nraw, vm, intr, dflatU, N);
  proj_acc_k<<<gp, 256, 0, stream>>>(coords, origin, nraw, vm, intr, dflatU, norms, ncnt, N);

  dim3 gi(HW / 256, NV);
  finalize_k<<<gi, 256, 0, stream>>>(dflatU, ncnt, norms, dt, dminb, dmaxb, tminb, tmaxb, cnt);
  norm_loss_k<<<gi, 256, 0, stream>>>(depths, dt, dtgt, dminb, dmaxb, tminb, tmaxb, lsum);
  loss_k<<<1, 32, 0, stream>>>(lsum, cnt, loss);
}